// Patcher_29197187678860
// MI455X (gfx1250) — compile-verified
//
#include <hip/hip_runtime.h>
#include <math.h>

// ---------------------------------------------------------------------------
// MI455X (gfx1250) implementation. All GEMMs use V_WMMA_F32_16X16X4_F32
// (fp32 in/out -> matches the fp32 reference numerics; the model is HBM-bound
// at 23.3 TB/s so fp32 matrix ops are the right precision/bandwidth tradeoff).
// Tiles are staged global->LDS with GLOBAL_LOAD_ASYNC_TO_LDS_B128 (ASYNCcnt)
// in a double-buffered pipeline.
// ---------------------------------------------------------------------------

typedef float v2f __attribute__((ext_vector_type(2)));
typedef float v8f __attribute__((ext_vector_type(8)));
typedef int   v4i __attribute__((vector_size(16)));

#define NTHREADS 256

#if __has_builtin(__builtin_amdgcn_global_load_async_to_lds_b128) && \
    __has_builtin(__builtin_amdgcn_s_wait_asynccnt)
#define HAVE_ASYNC_LDS 1
typedef __attribute__((address_space(1))) const void* gas_cvptr;
typedef __attribute__((address_space(1))) v4i*        gas_v4p;
typedef __attribute__((address_space(3))) void*       las_vptr;
typedef __attribute__((address_space(3))) v4i*        las_v4p;
// 16B async DMA: global (addrspace 1) -> LDS (addrspace 3), tracked by ASYNCcnt.
#define ASYNC_CP16(gsrc, ldst)                                        \
    __builtin_amdgcn_global_load_async_to_lds_b128(                   \
        (gas_v4p)(gas_cvptr)(gsrc), (las_v4p)(las_vptr)(ldst), 0, 0)
#else
#define HAVE_ASYNC_LDS 0
#endif

__device__ __forceinline__ float gelu_exact(float v) {
    return 0.5f * v * (1.0f + erff(v * 0.7071067811865475f));
}

// ------------------------------- GEMM (WMMA) -------------------------------
// C[M,N] = act(A[M,K] @ W[N,K]^T + bias[N]) (+ R[M,N])
// Block: 256 threads (8 waves), tile 64(M) x 128(N), K-stage 32.
// Wave grid 2x4; each wave owns a 32x32 patch = 2x2 v8f accumulators.
#define BM 64
#define BN 128
#define BK 32
#define LDT 36  // padded row stride (floats): 16B-aligned rows, conflict-free

__global__ __launch_bounds__(NTHREADS)
void gemm_wmma_kernel(const float* __restrict__ A, const float* __restrict__ W,
                      const float* __restrict__ bias, const float* __restrict__ R,
                      float* __restrict__ C,
                      int M, int N, int K, int doGelu, int hasBias, int hasRes) {
    __shared__ float As[2][BM][LDT];
    __shared__ float Bs[2][BN][LDT];
    const int tid  = threadIdx.x;
    const int lane = tid & 31;
    const int wid  = tid >> 5;
    const int bm = blockIdx.y * BM;
    const int bn = blockIdx.x * BN;
    const int wm = (wid & 1) * 32;   // wave row offset in block tile
    const int wn = (wid >> 1) * 32;  // wave col offset
    const int l16  = lane & 15;
    const int half = lane >> 4;      // A frag: lanes 0-15 hold K pair 0-1, 16-31 hold 2-3

    v8f acc00 = {0.f,0.f,0.f,0.f,0.f,0.f,0.f,0.f};
    v8f acc01 = acc00, acc10 = acc00, acc11 = acc00;

    const int nk = K / BK;

#if HAVE_ASYNC_LDS
    // Issue one stage of async tile DMA (A: 512 quads, W: 1024 quads).
    #define ISSUE_STAGE(k0, buf)                                                   \
    do {                                                                           \
        _Pragma("unroll")                                                          \
        for (int q = tid; q < (BM * BK) / 4; q += NTHREADS) {                      \
            int row = q >> 3, k4 = (q & 7) * 4;                                    \
            int gm = bm + row; if (gm >= M) gm = M - 1;                            \
            ASYNC_CP16(A + (size_t)gm * K + (k0) + k4, &As[buf][row][k4]);         \
        }                                                                          \
        _Pragma("unroll")                                                          \
        for (int q = tid; q < (BN * BK) / 4; q += NTHREADS) {                      \
            int row = q >> 3, k4 = (q & 7) * 4;                                    \
            ASYNC_CP16(W + (size_t)(bn + row) * K + (k0) + k4, &Bs[buf][row][k4]); \
        }                                                                          \
    } while (0)

    ISSUE_STAGE(0, 0);
    for (int s = 0; s < nk; ++s) {
        const int buf = s & 1;
        __builtin_amdgcn_s_wait_asynccnt(0);  // own stage-s DMA done
        __syncthreads();                      // everyone's stage-s DMA visible,
                                              // everyone done reading buf^1
        if (s + 1 < nk) ISSUE_STAGE((s + 1) * BK, buf ^ 1);
#else
    for (int s = 0; s < nk; ++s) {
        const int buf = 0;
        const int k0s = s * BK;
        __syncthreads();  // previous compute done before overwrite
        for (int idx = tid; idx < BM * BK; idx += NTHREADS) {
            int m = idx >> 5, k = idx & 31;
            int gm = bm + m; if (gm >= M) gm = M - 1;
            As[0][m][k] = A[(size_t)gm * K + k0s + k];
        }
        for (int idx = tid; idx < BN * BK; idx += NTHREADS) {
            int n = idx >> 5, k = idx & 31;
            Bs[0][n][k] = W[(size_t)(bn + n) * K + k0s + k];
        }
        __syncthreads();
#endif
#pragma unroll
        for (int kk = 0; kk < BK; kk += 4) {
            const int kf = kk + half * 2;
            v2f a0 = *(const v2f*)&As[buf][wm + l16][kf];
            v2f a1 = *(const v2f*)&As[buf][wm + 16 + l16][kf];
            v2f b0 = *(const v2f*)&Bs[buf][wn + l16][kf];
            v2f b1 = *(const v2f*)&Bs[buf][wn + 16 + l16][kf];
            acc00 = __builtin_amdgcn_wmma_f32_16x16x4_f32(false, a0, false, b0, (short)0, acc00, false, false);
            acc01 = __builtin_amdgcn_wmma_f32_16x16x4_f32(false, a0, false, b1, (short)0, acc01, false, false);
            acc10 = __builtin_amdgcn_wmma_f32_16x16x4_f32(false, a1, false, b0, (short)0, acc10, false, false);
            acc11 = __builtin_amdgcn_wmma_f32_16x16x4_f32(false, a1, false, b1, (short)0, acc11, false, false);
        }
    }

    // Epilogue: D layout — vgpr j: lanes 0-15 -> row base+j, lanes 16-31 -> row base+8+j.
#pragma unroll
    for (int ms = 0; ms < 2; ++ms) {
#pragma unroll
        for (int j = 0; j < 8; ++j) {
            int row = bm + wm + ms * 16 + half * 8 + j;
            if (row >= M) continue;
            int c0 = bn + wn + l16;
            int c1 = c0 + 16;
            float v0 = ms ? acc10[j] : acc00[j];
            float v1 = ms ? acc11[j] : acc01[j];
            if (hasBias) { v0 += bias[c0]; v1 += bias[c1]; }
            if (doGelu)  { v0 = gelu_exact(v0); v1 = gelu_exact(v1); }
            size_t o0 = (size_t)row * N + c0;
            size_t o1 = (size_t)row * N + c1;
            if (hasRes) { v0 += R[o0]; v1 += R[o1]; }
            C[o0] = v0;
            C[o1] = v1;
        }
    }
}

// ------------------------------- LayerNorm ---------------------------------
__global__ __launch_bounds__(NTHREADS)
void layernorm_kernel(const float* __restrict__ x, const float* __restrict__ w,
                      float* __restrict__ out, int C) {
    __shared__ float red[NTHREADS];
    const size_t row = blockIdx.x;
    const float* xr = x + row * (size_t)C;
    float* orow = out + row * (size_t)C;
    const int tid = threadIdx.x;
    float s = 0.f;
    for (int c = tid; c < C; c += NTHREADS) s += xr[c];
    red[tid] = s; __syncthreads();
    for (int sh = 128; sh > 0; sh >>= 1) { if (tid < sh) red[tid] += red[tid + sh]; __syncthreads(); }
    float mean = red[0] / (float)C; __syncthreads();
    float s2 = 0.f;
    for (int c = tid; c < C; c += NTHREADS) { float d = xr[c] - mean; s2 += d * d; }
    red[tid] = s2; __syncthreads();
    for (int sh = 128; sh > 0; sh >>= 1) { if (tid < sh) red[tid] += red[tid + sh]; __syncthreads(); }
    float inv = rsqrtf(red[0] / (float)C + 1e-5f); __syncthreads();
    for (int c = tid; c < C; c += NTHREADS) orow[c] = (xr[c] - mean) * inv * w[c];
}

// ------------------------------ Row softmax --------------------------------
__global__ __launch_bounds__(NTHREADS)
void rowsoftmax_kernel(const float* __restrict__ in, float* __restrict__ out, int n) {
    __shared__ float red[NTHREADS];
    const size_t row = blockIdx.x;
    const float* ir = in + row * (size_t)n;
    float* orow = out + row * (size_t)n;
    const int tid = threadIdx.x;
    float m = -INFINITY;
    for (int i = tid; i < n; i += NTHREADS) m = fmaxf(m, ir[i]);
    red[tid] = m; __syncthreads();
    for (int sh = 128; sh > 0; sh >>= 1) { if (tid < sh) red[tid] = fmaxf(red[tid], red[tid + sh]); __syncthreads(); }
    m = red[0]; __syncthreads();
    float s = 0.f;
    for (int i = tid; i < n; i += NTHREADS) s += expf(ir[i] - m);
    red[tid] = s; __syncthreads();
    for (int sh = 128; sh > 0; sh >>= 1) { if (tid < sh) red[tid] += red[tid + sh]; __syncthreads(); }
    float inv = 1.f / red[0]; __syncthreads();
    for (int i = tid; i < n; i += NTHREADS) orow[i] = expf(ir[i] - m) * inv;
}

// ------------------------------- Attention ---------------------------------
// One block per (query row, head, batch). Scores in LDS, two-pass softmax,
// coalesced V accumulation. Handles causal (D-blocks) and cross (sblock).
__global__ __launch_bounds__(NTHREADS)
void attn_kernel(const float* __restrict__ Q, const float* __restrict__ K,
                 const float* __restrict__ V, float* __restrict__ Y,
                 int Tq, int Tk, int hd, int qstride, int kvstride, int ystride,
                 float scale, int causal) {
    __shared__ float qs[512];
    __shared__ float sc[1024];
    __shared__ float red[NTHREADS];
    const int tq = blockIdx.x, h = blockIdx.y, b = blockIdx.z;
    const int tid = threadIdx.x;
    const float* qrow = Q + ((size_t)b * Tq + tq) * qstride + (size_t)h * hd;
    for (int d = tid; d < hd; d += NTHREADS) qs[d] = qrow[d];
    __syncthreads();
    const int jmax = causal ? (tq + 1) : Tk;
    for (int j = tid; j < jmax; j += NTHREADS) {
        const float* krow = K + ((size_t)b * Tk + j) * kvstride + (size_t)h * hd;
        float acc = 0.f;
        for (int d = 0; d < hd; ++d) acc += qs[d] * krow[d];
        sc[j] = acc * scale;
    }
    __syncthreads();
    float lm = -INFINITY;
    for (int j = tid; j < jmax; j += NTHREADS) lm = fmaxf(lm, sc[j]);
    red[tid] = lm; __syncthreads();
    for (int sh = 128; sh > 0; sh >>= 1) { if (tid < sh) red[tid] = fmaxf(red[tid], red[tid + sh]); __syncthreads(); }
    float m = red[0]; __syncthreads();
    float ls = 0.f;
    for (int j = tid; j < jmax; j += NTHREADS) { float e = expf(sc[j] - m); sc[j] = e; ls += e; }
    red[tid] = ls; __syncthreads();
    for (int sh = 128; sh > 0; sh >>= 1) { if (tid < sh) red[tid] += red[tid + sh]; __syncthreads(); }
    float inv = 1.f / red[0]; __syncthreads();
    for (int d = tid; d < hd; d += NTHREADS) {
        float acc = 0.f;
        for (int j = 0; j < jmax; ++j)
            acc += sc[j] * V[((size_t)b * Tk + j) * kvstride + (size_t)h * hd + d];
        Y[((size_t)b * Tq + tq) * ystride + (size_t)h * hd + d] = acc * inv;
    }
}

// ------------------------------ Gate (sumsq) -------------------------------
__global__ __launch_bounds__(NTHREADS)
void gate_kernel(const float* __restrict__ x, const float* __restrict__ scan,
                 float* __restrict__ gate, int T, int C) {
    __shared__ float red[NTHREADS];
    const int t = blockIdx.x % T, b = blockIdx.x / T;
    const int tid = threadIdx.x;
    float s = 0.f;
    if (t < T - 1) {
        const float* xr = x + ((size_t)b * T + t + 1) * C;
        const float* sr = scan + ((size_t)b * T + t) * C;
        for (int c = tid; c < C; c += NTHREADS) { float d = xr[c] - sr[c]; s += d * d; }
    }
    red[tid] = s; __syncthreads();
    for (int sh = 128; sh > 0; sh >>= 1) { if (tid < sh) red[tid] += red[tid + sh]; __syncthreads(); }
    if (tid == 0) gate[blockIdx.x] = red[0];
}

// -------------------------- Top-k (stable argmax) --------------------------
__global__ __launch_bounds__(NTHREADS)
void topk_kernel(const float* __restrict__ g, float* __restrict__ outv,
                 int* __restrict__ outi, int n, int k) {
    __shared__ float vals[1024];
    __shared__ float rv[NTHREADS];
    __shared__ int   ri[NTHREADS];
    const int b = blockIdx.x, tid = threadIdx.x;
    for (int i = tid; i < n; i += NTHREADS) vals[i] = g[(size_t)b * n + i];
    __syncthreads();
    for (int sel = 0; sel < k; ++sel) {
        float bv = -INFINITY; int bi = n;
        for (int i = tid; i < n; i += NTHREADS) {
            float v = vals[i];
            if (v > bv || (v == bv && i < bi)) { bv = v; bi = i; }
        }
        rv[tid] = bv; ri[tid] = bi; __syncthreads();
        for (int sh = 128; sh > 0; sh >>= 1) {
            if (tid < sh) {
                float ov = rv[tid + sh]; int oi = ri[tid + sh];
                if (ov > rv[tid] || (ov == rv[tid] && oi < ri[tid])) { rv[tid] = ov; ri[tid] = oi; }
            }
            __syncthreads();
        }
        if (tid == 0) {
            outv[(size_t)b * k + sel] = rv[0];
            outi[(size_t)b * k + sel] = ri[0];
            vals[ri[0]] = -INFINITY;
        }
        __syncthreads();
    }
}

// ------------------------------ Gather/Scatter -----------------------------
__global__ __launch_bounds__(NTHREADS)
void gather_kernel(const float* __restrict__ sm, const int* __restrict__ idx,
                   const float* __restrict__ w, float* __restrict__ out,
                   int T, int C, int K) {
    const int b = blockIdx.x / K, i = blockIdx.x % K;
    const int src = idx[(size_t)b * K + i];
    const float wv = w[(size_t)b * K + i];
    const float* s = sm + ((size_t)b * T + src) * C;
    float* o = out + ((size_t)b * K + i) * C;
    for (int c = threadIdx.x; c < C; c += NTHREADS) o[c] = s[c] * wv;
}

__global__ __launch_bounds__(NTHREADS)
void flatten_kernel(const float* __restrict__ passed, float* __restrict__ flat,
                    int K, int C, int S) {
    const int b = blockIdx.x;
    for (int j = threadIdx.x; j < K * S; j += NTHREADS) {
        int i = j / S, s = j % S;
        flat[(size_t)b * K * S + j] = passed[((size_t)b * K + i) * C + s];
    }
}

// scattered[b,t,c] = pos_w[c,t] + pos_b[c]   (scatter adds on top; order irrelevant)
__global__ __launch_bounds__(NTHREADS)
void scatter_init_kernel(float* __restrict__ scat, const float* __restrict__ pos_w,
                         const float* __restrict__ pos_b, int T, int C) {
    const int t = blockIdx.x % T; const size_t b = blockIdx.x / T;
    float* o = scat + (b * T + t) * (size_t)C;
    for (int c = threadIdx.x; c < C; c += NTHREADS) o[c] = pos_w[(size_t)c * T + t] + pos_b[c];
}

__global__ __launch_bounds__(NTHREADS)
void scatter_add_kernel(const float* __restrict__ passed, const int* __restrict__ sidx,
                        const float* __restrict__ sw, float* __restrict__ scat,
                        int T, int C, int K) {
    const int b = blockIdx.x / K, i = blockIdx.x % K;
    const int dst = sidx[(size_t)b * K + i];
    const float wv = sw[(size_t)b * K + i];
    const float* p = passed + ((size_t)b * K + i) * C;
    float* o = scat + ((size_t)b * T + dst) * C;
    for (int c = threadIdx.x; c < C; c += NTHREADS) o[c] += p[c] * wv;
}

__global__ void write_loss_kernel(float* p) { *p = 0.f; }

// ---------------------------------------------------------------------------
static inline void gemm(hipStream_t st, const float* A, const float* W, const float* bias,
                        const float* R, float* C, int M, int N, int K,
                        int doGelu, int hasBias, int hasRes) {
    dim3 grid(N / BN, (M + BM - 1) / BM);
    gemm_wmma_kernel<<<grid, NTHREADS, 0, st>>>(A, W, bias, R, C, M, N, K, doGelu, hasBias, hasRes);
}

extern "C" void kernel_launch(void* const* d_in, const int* in_sizes, int n_in,
                              void* d_out, int out_size, void* d_ws, size_t ws_size,
                              hipStream_t stream) {
    (void)in_sizes; (void)n_in; (void)out_size; (void)ws_size;
    constexpr int Bb = 8, T = 1024, D = 1024, I = 2048, H = 4, IS = 256, SD = 10;

    int a = 0;
    const float* x          = (const float*)d_in[a++];
    const float* us_ln1     = (const float*)d_in[a++];
    const float* us_attn_w  = (const float*)d_in[a++];
    const float* us_proj_w  = (const float*)d_in[a++];
    const float* us_ln2     = (const float*)d_in[a++];
    const float* us_fc_w    = (const float*)d_in[a++];
    const float* us_mproj_w = (const float*)d_in[a++];
    const float* up_proj_w  = (const float*)d_in[a++];
    const float* up_proj_b  = (const float*)d_in[a++];
    const float* pos_w      = (const float*)d_in[a++];
    const float* pos_b      = (const float*)d_in[a++];
    const float* sg_w       = (const float*)d_in[a++];
    const float* sg_b       = (const float*)d_in[a++];
    const float* sb_ln1     = (const float*)d_in[a++];
    const float* sb_ln2     = (const float*)d_in[a++];
    const float* sb_ln3     = (const float*)d_in[a++];
    const float* sb_q_w     = (const float*)d_in[a++];
    const float* sb_k_w     = (const float*)d_in[a++];
    const float* sb_v_w     = (const float*)d_in[a++];
    const float* sb_c_w     = (const float*)d_in[a++];
    const float* sb_fc_w    = (const float*)d_in[a++];
    const float* sb_mproj_w = (const float*)d_in[a++];
    const float* down_proj_w= (const float*)d_in[a++];
    const float* down_proj_b= (const float*)d_in[a++];
    const float* ds_ln1     = (const float*)d_in[a++];
    const float* ds_attn_w  = (const float*)d_in[a++];
    const float* ds_proj_w  = (const float*)d_in[a++];
    const float* ds_ln2     = (const float*)d_in[a++];
    const float* ds_fc_w    = (const float*)d_in[a++];
    const float* ds_mproj_w = (const float*)d_in[a++];

    float* ws = (float*)d_ws;
    size_t off = 0;
    auto alloc = [&](size_t n) { float* p = ws + off; off += n; return p; };
    float* scanb  = alloc((size_t)Bb * T * D);        // 32MB  scan (persists)
    float* big1   = alloc((size_t)Bb * T * 3 * D);    // 96MB  qkv / q
    float* big2   = alloc((size_t)4096 * 4096);       // 64MB  MLP chunk buffer
    float* x1b    = alloc((size_t)Bb * T * D);        // 32MB  x1 / k+v / z
    float* yb     = alloc((size_t)Bb * T * D);        // 32MB  attn y / gathered / ln_p
    float* lnb    = alloc((size_t)Bb * T * I);        // 64MB  LN scratch / sm_scan / y_s
    float* passed = alloc((size_t)Bb * IS * I);       // 16MB
    float* scat   = alloc((size_t)Bb * T * I);        // 64MB  scattered / sx1 / pds
    float* gateb  = alloc((size_t)Bb * T);
    float* gb     = alloc((size_t)Bb * T);
    float* topv   = alloc((size_t)Bb * IS);
    int*   topi   = (int*)alloc((size_t)Bb * IS);
    float* flatb  = alloc((size_t)Bb * IS * SD);
    float* sgl    = alloc((size_t)Bb * T);
    float* sgs    = alloc((size_t)Bb * T);
    float* swv    = alloc((size_t)Bb * IS);
    int*   swi    = (int*)alloc((size_t)Bb * IS);
    float* out    = (float*)d_out;

    const int rowsBT = Bb * T;   // 8192
    const int rowsBI = Bb * IS;  // 2048
    const int hdD = D / H;       // 256
    const int hdI = I / H;       // 512

    // -------- upstream block --------
    layernorm_kernel<<<rowsBT, NTHREADS, 0, stream>>>(x, us_ln1, lnb, D);
    gemm(stream, lnb, us_attn_w, nullptr, nullptr, big1, rowsBT, 3 * D, D, 0, 0, 0);
    attn_kernel<<<dim3(T, H, Bb), NTHREADS, 0, stream>>>(big1, big1 + D, big1 + 2 * D, yb,
                                                         T, T, hdD, 3 * D, 3 * D, D, 0.0625f, 1);
    gemm(stream, yb, us_proj_w, nullptr, x, x1b, rowsBT, D, D, 0, 0, 1);
    layernorm_kernel<<<rowsBT, NTHREADS, 0, stream>>>(x1b, us_ln2, lnb, D);
    for (int c = 0; c < 2; ++c) {
        size_t r0 = (size_t)c * 4096;
        gemm(stream, lnb + r0 * D, us_fc_w, nullptr, nullptr, big2, 4096, 4 * D, D, 1, 0, 0);
        gemm(stream, big2, us_mproj_w, nullptr, x1b + r0 * D, scanb + r0 * D, 4096, D, 4 * D, 0, 0, 1);
    }

    // -------- gate + gather + up-proj --------
    gate_kernel<<<rowsBT, NTHREADS, 0, stream>>>(x, scanb, gateb, T, D);
    rowsoftmax_kernel<<<Bb, NTHREADS, 0, stream>>>(gateb, gb, T);
    rowsoftmax_kernel<<<rowsBT, NTHREADS, 0, stream>>>(scanb, lnb, D);      // sm_scan
    topk_kernel<<<Bb, NTHREADS, 0, stream>>>(gb, topv, topi, T, IS);
    gather_kernel<<<Bb * IS, NTHREADS, 0, stream>>>(lnb, topi, topv, yb, T, D, IS);
    gemm(stream, yb, up_proj_w, up_proj_b, nullptr, passed, rowsBI, I, D, 0, 1, 0);

    // -------- sg gate + scatter --------
    flatten_kernel<<<Bb, NTHREADS, 0, stream>>>(passed, flatb, IS, I, SD);
    gemm(stream, flatb, sg_w, sg_b, nullptr, sgl, Bb, T, IS * SD, 0, 1, 0);
    rowsoftmax_kernel<<<Bb, NTHREADS, 0, stream>>>(sgl, sgs, T);
    topk_kernel<<<Bb, NTHREADS, 0, stream>>>(sgs, swv, swi, T, IS);
    scatter_init_kernel<<<Bb * T, NTHREADS, 0, stream>>>(scat, pos_w, pos_b, T, I);
    scatter_add_kernel<<<Bb * IS, NTHREADS, 0, stream>>>(passed, swi, swv, scat, T, I, IS);

    // -------- sblock (cross attention, I=2048) --------
    layernorm_kernel<<<rowsBI, NTHREADS, 0, stream>>>(passed, sb_ln1, yb, I);   // ln(x)
    layernorm_kernel<<<rowsBT, NTHREADS, 0, stream>>>(scat, sb_ln2, lnb, I);    // ln(sx)
    gemm(stream, lnb, sb_q_w, nullptr, nullptr, big1, rowsBT, I, I, 0, 0, 0);   // q
    gemm(stream, yb, sb_k_w, nullptr, nullptr, x1b, rowsBI, I, I, 0, 0, 0);     // k
    gemm(stream, yb, sb_v_w, nullptr, nullptr, x1b + (size_t)Bb * IS * I, rowsBI, I, I, 0, 0, 0); // v
    attn_kernel<<<dim3(T, H, Bb), NTHREADS, 0, stream>>>(big1, x1b, x1b + (size_t)Bb * IS * I,
                                                         lnb, T, IS, hdI, I, I, I, 1.0f, 0);
    gemm(stream, lnb, sb_c_w, nullptr, scat, scat, rowsBT, I, I, 0, 0, 1);      // sx1 (in-place)
    layernorm_kernel<<<rowsBT, NTHREADS, 0, stream>>>(scat, sb_ln3, lnb, I);
    for (int c = 0; c < 4; ++c) {
        size_t r0 = (size_t)c * 2048;
        gemm(stream, lnb + r0 * I, sb_fc_w, nullptr, nullptr, big2, 2048, 4 * I, I, 1, 0, 0);
        gemm(stream, big2, sb_mproj_w, nullptr, scat + r0 * I, scat + r0 * I, 2048, I, 4 * I, 0, 0, 1);
    }
    // z = pds @ down^T + b + scan
    gemm(stream, scat, down_proj_w, down_proj_b, scanb, x1b, rowsBT, D, I, 0, 1, 1);

    // -------- downstream block --------
    layernorm_kernel<<<rowsBT, NTHREADS, 0, stream>>>(x1b, ds_ln1, lnb, D);
    gemm(stream, lnb, ds_attn_w, nullptr, nullptr, big1, rowsBT, 3 * D, D, 0, 0, 0);
    attn_kernel<<<dim3(T, H, Bb), NTHREADS, 0, stream>>>(big1, big1 + D, big1 + 2 * D, yb,
                                                         T, T, hdD, 3 * D, 3 * D, D, 0.0625f, 1);
    gemm(stream, yb, ds_proj_w, nullptr, x1b, scanb, rowsBT, D, D, 0, 0, 1);    // x1d
    layernorm_kernel<<<rowsBT, NTHREADS, 0, stream>>>(scanb, ds_ln2, lnb, D);
    for (int c = 0; c < 2; ++c) {
        size_t r0 = (size_t)c * 4096;
        gemm(stream, lnb + r0 * D, ds_fc_w, nullptr, nullptr, big2, 4096, 4 * D, D, 1, 0, 0);
        gemm(stream, big2, ds_mproj_w, nullptr, scanb + r0 * D, out + r0 * D, 4096, D, 4 * D, 0, 0, 1);
    }
    write_loss_kernel<<<1, 1, 0, stream>>>(out + (size_t)Bb * T * D);
}